// MZIBlockLinear_360777253008
// MI455X (gfx1250) — compile-verified
//
#include <hip/hip_runtime.h>

// ---------------------------------------------------------------------------
// MZIBlockLinear for MI455X (gfx1250, wave32, WMMA)
//
//   Phase 1: W[p*8+i][q*8+j] = sum_k u[p,q,i,k]*s[p,q,k]*v[p,q,k,j]
//            (plain row-major W -> phase-2 A and B tiles have identical
//             geometry: 128 rows x 32 contiguous floats)   ~0.27 GFLOP
//   Phase 2: out[4096,4096] = x @ W^T                      ~137 GFLOP fp32
//            compute-bound => native fp32 WMMA (V_WMMA_F32_16X16X4_F32),
//            double-buffered LDS staged via GLOBAL_LOAD_ASYNC_TO_LDS_B128
//            (ASYNCcnt) when available, register prefetch otherwise.
// ---------------------------------------------------------------------------

typedef float v2f __attribute__((ext_vector_type(2)));
typedef float v8f __attribute__((ext_vector_type(8)));
typedef int   v4i_ __attribute__((vector_size(4 * sizeof(int))));  // b128 payload

#if __has_builtin(__builtin_amdgcn_global_load_async_to_lds_b128)
#define USE_ASYNC_LDS 1
#else
#define USE_ASYNC_LDS 0
#endif

#define P_DIM 512
#define Q_DIM 512
#define N_DIM 4096   // P*8  (out cols)
#define K_DIM 4096   // Q*8  (reduction)
#define M_DIM 4096   // batch (out rows)

// -------------------------------- Phase 1 ----------------------------------
__global__ __launch_bounds__(64) void build_w_kernel(
    const float* __restrict__ u, const float* __restrict__ s,
    const float* __restrict__ v, float* __restrict__ w)
{
    __shared__ float su[64];
    __shared__ float sv[64];
    __shared__ float ss[8];

    const int pq = blockIdx.x;           // p*Q + q
    const int t  = threadIdx.x;          // 0..63
    const size_t base = (size_t)pq * 64;

    su[t] = u[base + t];
    sv[t] = v[base + t];
    if (t < 8) ss[t] = s[(size_t)pq * 8 + t];
    __syncthreads();

    const int i = t >> 3, j = t & 7;
    float acc = 0.f;
#pragma unroll
    for (int k = 0; k < 8; ++k)
        acc += su[i * 8 + k] * ss[k] * sv[k * 8 + j];

    const int p = pq >> 9;               // pq / 512
    const int q = pq & 511;              // pq % 512
    // row-major W[n][kdim]: n = p*8+i, kdim = q*8+j
    w[(size_t)(p * 8 + i) * K_DIM + (q * 8 + j)] = acc;
}

// -------------------------------- Phase 2 ----------------------------------
// 256 threads = 8 waves, 128x128 output tile, K-chunk 32.
// LDS: As[m][k] and Bs[n][k], row stride 36 floats (36 ≡ 4 mod 32, distinct
// multiples of 4 per row mod 64 -> conflict-free float2 fragment loads).
// Both tiles are 128 rows x 32 floats staged as 8 b128 copies per thread.
#define MT   128
#define NT   128
#define KT   32
#define LSTR 36
#define ATILE (MT * LSTR)            // 4608 floats
#define BTILE (NT * LSTR)            // 4608 floats
#define BUFSZ (ATILE + BTILE)        // floats per buffer
#define LDS_BYTES (2 * BUFSZ * 4)    // 73,728 B (double buffered; 4 WGs/WGP)

#if USE_ASYNC_LDS
__device__ __forceinline__ void stage_tile_async(const float* gbase, float* lbase,
                                                 int tid) {
#pragma unroll
    for (int i = 0; i < 4; ++i) {
        const int idx = tid + i * 256;
        const int r = idx >> 3, c = idx & 7;   // 128 rows x 8 float4
        const float* g = gbase + (size_t)r * K_DIM + c * 4;
        float* l = lbase + r * LSTR + c * 4;
        __builtin_amdgcn_global_load_async_to_lds_b128(
            (v4i_*)g, (v4i_*)l, /*offset=*/0, /*cpol=*/0);
    }
}
__device__ __forceinline__ void wait_async_done() {
#if __has_builtin(__builtin_amdgcn_s_wait_asynccnt)
    __builtin_amdgcn_s_wait_asynccnt(0);
#else
    asm volatile("s_wait_asynccnt 0" ::: "memory");
#endif
}
#else
__device__ __forceinline__ void stage_tile_ldst(const float* gbase, float* lbase,
                                                int tid) {
    float4 rr[4];
#pragma unroll
    for (int i = 0; i < 4; ++i) {
        const int idx = tid + i * 256;
        const int r = idx >> 3, c = idx & 7;
        rr[i] = *(const float4*)(gbase + (size_t)r * K_DIM + c * 4);
    }
#pragma unroll
    for (int i = 0; i < 4; ++i) {
        const int idx = tid + i * 256;
        const int r = idx >> 3, c = idx & 7;
        *(float4*)(lbase + r * LSTR + c * 4) = rr[i];
    }
}
#endif

__global__ __launch_bounds__(256) void gemm_wmma_f32_kernel(
    const float* __restrict__ x, const float* __restrict__ w,
    float* __restrict__ out)
{
    extern __shared__ __align__(16) float lds[];   // [2][ATILE + BTILE]

    const int tid  = threadIdx.x;
    const int wave = tid >> 5;
    const int lane = tid & 31;
    const int l16  = lane & 15;
    const int khi  = (lane >> 4) * 2;   // lanes 16-31 hold K+2,K+3 (16x4 A layout)

    const int m0 = blockIdx.y * MT;
    const int n0 = blockIdx.x * NT;

    const float* xa = x + (size_t)m0 * K_DIM;   // A tile base (rows = batch)
    const float* wb = w + (size_t)n0 * K_DIM;   // B tile base (rows = out col)

    v8f acc[8] = {};                    // 8 x (16x16 f32) accumulators

    // ---- stage K-tile 0 into buffer 0 ----
#if USE_ASYNC_LDS
    stage_tile_async(xa, lds, tid);
    stage_tile_async(wb, lds + ATILE, tid);
    wait_async_done();
#else
    stage_tile_ldst(xa, lds, tid);
    stage_tile_ldst(wb, lds + ATILE, tid);
#endif
    __syncthreads();

    int cur = 0;
    for (int kb = 0; kb < K_DIM; kb += KT) {
        const int kn = kb + KT;
        float* nxt = lds + (cur ^ 1) * BUFSZ;

        // issue next tile's copies before the WMMAs (overlap DMA w/ math)
        if (kn < K_DIM) {
#if USE_ASYNC_LDS
            stage_tile_async(xa + kn, nxt, tid);
            stage_tile_async(wb + kn, nxt + ATILE, tid);
#endif
        }

        // ---- 8 k-steps of 4, 8 N-subtiles: 64 WMMAs per wave per chunk ----
        const float* As = lds + cur * BUFSZ;
        const float* Bs = As + ATILE;
#pragma unroll
        for (int k4 = 0; k4 < 8; ++k4) {
            const v2f a = *(const v2f*)(As + (wave * 16 + l16) * LSTR + k4 * 4 + khi);
#pragma unroll
            for (int nt = 0; nt < 8; ++nt) {
                const v2f b = *(const v2f*)(Bs + (nt * 16 + l16) * LSTR + k4 * 4 + khi);
                acc[nt] = __builtin_amdgcn_wmma_f32_16x16x4_f32(
                    /*neg_a=*/false, a, /*neg_b=*/false, b,
                    /*c_mod=*/(short)0, acc[nt],
                    /*reuse_a=*/false, /*reuse_b=*/false);
            }
        }

        if (kn < K_DIM) {
#if USE_ASYNC_LDS
            wait_async_done();          // this wave's DMA into nxt has landed
#else
            stage_tile_ldst(xa + kn, nxt, tid);
            stage_tile_ldst(wb + kn, nxt + ATILE, tid);
#endif
        }
        __syncthreads();                // all waves done reading cur / writing nxt
        cur ^= 1;
    }

    // ---- epilogue: 16x16 f32 C/D layout ----
    // VGPR r: lanes 0-15 -> M=r, lanes 16-31 -> M=r+8; N = lane&15
    const int rowb = m0 + wave * 16 + (lane >> 4) * 8;
#pragma unroll
    for (int nt = 0; nt < 8; ++nt) {
        const int col = n0 + nt * 16 + l16;
#pragma unroll
        for (int r = 0; r < 8; ++r) {
            out[(size_t)(rowb + r) * N_DIM + col] = acc[nt][r];
        }
    }
}

// ---------------------------------------------------------------------------
extern "C" void kernel_launch(void* const* d_in, const int* in_sizes, int n_in,
                              void* d_out, int out_size, void* d_ws, size_t ws_size,
                              hipStream_t stream) {
    (void)in_sizes; (void)n_in; (void)out_size; (void)ws_size;
    const float* x = (const float*)d_in[0];   // [4096, 4096]
    const float* u = (const float*)d_in[1];   // [512, 512, 8, 8]
    const float* s = (const float*)d_in[2];   // [512, 512, 8]
    const float* v = (const float*)d_in[3];   // [512, 512, 8, 8]
    float* out = (float*)d_out;               // [4096, 4096]
    float* w   = (float*)d_ws;                // 4096*4096 floats = 64 MB scratch

    // Phase 1: build row-major W[n][k]
    build_w_kernel<<<dim3(P_DIM * Q_DIM), dim3(64), 0, stream>>>(u, s, v, w);

    // Phase 2: out = x @ W^T  (4096^3 fp32 GEMM via v_wmma_f32_16x16x4_f32)
    gemm_wmma_f32_kernel<<<dim3(N_DIM / NT, M_DIM / MT), dim3(256), LDS_BYTES, stream>>>(
        x, w, out);
}